// MultiheadAttention_12756052869168
// MI455X (gfx1250) — compile-verified
//
#include <hip/hip_runtime.h>
#include <stdint.h>

typedef float v2f __attribute__((ext_vector_type(2)));
typedef float v8f __attribute__((ext_vector_type(8)));

#define BH 64
#define T 1024
#define D 64
#define QT 16                      // query rows per block (WMMA M)
#define NWAVE 4
#define NTHREADS (NWAVE * 32)
#define COLS_PER_WAVE (T / NWAVE)  // 256
#define KTILES (COLS_PER_WAVE / 16)

#define NEG_FILL (-4294967295.0f)

// XOR-swizzle LDS columns by row so WMMA A-fragment reads (fixed column,
// varying row) spread across banks instead of serializing 16-way.
__device__ __forceinline__ int swz(int m, int c) { return c ^ ((m & 7) << 3); }

__global__ __launch_bounds__(NTHREADS)
void mha_fwd_kernel(const float* __restrict__ key,
                    const float* __restrict__ value,
                    const float* __restrict__ query,
                    const uint8_t* __restrict__ mask,
                    const float* __restrict__ qmask,
                    const float* __restrict__ gauss,
                    float* __restrict__ out_result,
                    float* __restrict__ out_attn)
{
  __shared__ float sS[QT][T];          // 64KB: scores -> exp -> probabilities
  __shared__ float sWmax[NWAVE][QT];   // per-wave row maxima
  __shared__ float sWsum[NWAVE][QT];   // per-wave row exp-sums
  __shared__ float sInv[QT];           // 1 / rowsum

  const int bh = blockIdx.x / (T / QT);
  const int qt = blockIdx.x % (T / QT);
  const int q0 = qt * QT;
  const int tid  = threadIdx.x;
  const int wave = tid >> 5;
  const int lane = tid & 31;

  const float*   Qb = query + ((size_t)bh * T + q0) * D;
  const float*   Kb = key   + (size_t)bh * T * D;
  const float*   Vb = value + (size_t)bh * T * D;
  const float*   Gb = gauss + ((size_t)bh * T + q0) * T;
  const uint8_t* Mb = mask  + ((size_t)bh * T + q0) * T;
  const float*   Pb = qmask + ((size_t)bh * T + q0) * T;
  float* attn_out = out_attn + ((size_t)bh * T + q0) * T;

  const int fm   = lane & 15;   // A row / B,C column index for this lane
  const int half = lane >> 4;   // lane half selects K-pair / C row-base
  const int koff = half * 2;

  // ---- Q A-fragments (16x64 tile as 16 fragments of 16x4), pre-scaled ----
  const float scale = 0.125f;   // 1/sqrt(64)
  float qa0[16], qa1[16];
  #pragma unroll
  for (int f = 0; f < 16; ++f) {
    qa0[f] = Qb[fm * D + 4 * f + koff    ] * scale;
    qa1[f] = Qb[fm * D + 4 * f + koff + 1] * scale;
  }

  // ---- scores: S = (Q/sqrt(D)) K^T - gauss, with mask fill ----
  const int cw0 = wave * COLS_PER_WAVE;
  float rmax[8];
  #pragma unroll
  for (int r = 0; r < 8; ++r) rmax[r] = -INFINITY;

  for (int t = 0; t < KTILES; ++t) {
    const int k0 = cw0 + t * 16;
    const float* Krow = Kb + (size_t)(k0 + fm) * D + koff;
    if (t + 1 < KTILES)
      __builtin_prefetch(Kb + (size_t)(k0 + 16 + fm) * D + koff, 0, 0);
    v8f c = {};
    #pragma unroll
    for (int f = 0; f < 16; ++f) {
      v2f a, b;
      a.x = qa0[f];      a.y = qa1[f];
      b.x = Krow[4 * f]; b.y = Krow[4 * f + 1];
      c = __builtin_amdgcn_wmma_f32_16x16x4_f32(false, a, false, b,
                                                (short)0, c, false, false);
    }
    // C layout: VGPR r holds row m = r + 8*half, column n = fm
    const int n  = fm;
    const int mb = half * 8;
    #pragma unroll
    for (int r = 0; r < 8; ++r) {
      const int m = mb + r;
      float s = c[r] - Gb[(size_t)m * T + k0 + n];
      if (Mb[(size_t)m * T + k0 + n]) s = NEG_FILL;
      sS[m][swz(m, k0 + n)] = s;
      rmax[r] = fmaxf(rmax[r], s);
    }
  }

  // reduce row maxima across the 16 lanes (columns) of each half
  #pragma unroll
  for (int d2 = 1; d2 < 16; d2 <<= 1) {
    #pragma unroll
    for (int r = 0; r < 8; ++r)
      rmax[r] = fmaxf(rmax[r], __shfl_xor(rmax[r], d2, 32));
  }
  if (fm == 0) {
    #pragma unroll
    for (int r = 0; r < 8; ++r) sWmax[wave][half * 8 + r] = rmax[r];
  }
  __syncthreads();

  // ---- exp pass over this wave's 256-column strip; per-row sums ----
  {
    const int row = lane & 15;
    const int h   = lane >> 4;
    const float rm = fmaxf(fmaxf(sWmax[0][row], sWmax[1][row]),
                           fmaxf(sWmax[2][row], sWmax[3][row]));
    float sum = 0.0f;
    const int cbase = cw0 + h * 128;
    #pragma unroll 4
    for (int j = 0; j < 128; ++j) {
      const int c = cbase + j;
      const int pc = swz(row, c);
      float e = __expf(sS[row][pc] - rm);
      sS[row][pc] = e;
      sum += e;
    }
    sum += __shfl_xor(sum, 16, 32);
    if (h == 0) sWsum[wave][row] = sum;
  }
  __syncthreads();
  if (tid < QT)
    sInv[tid] = 1.0f / (sWsum[0][tid] + sWsum[1][tid] +
                        sWsum[2][tid] + sWsum[3][tid]);
  __syncthreads();

  // ---- normalize * query_mask; write attn (coalesced); keep P in LDS ----
  for (int i = tid; i < QT * T; i += NTHREADS) {
    const int row = i >> 10;
    const int col = i & (T - 1);
    const int pc  = swz(row, col);
    float p = sS[row][pc] * sInv[row] * Pb[(size_t)row * T + col];
    sS[row][pc] = p;
    attn_out[(size_t)row * T + col] = p;
  }
  __syncthreads();

  // ---- result = P @ V : each wave owns a 16-wide slice of D ----
  {
    const int n0 = wave * 16;
    v8f acc = {};
    #pragma unroll 8
    for (int kk = 0; kk < T; kk += 4) {
      v2f a, b;
      a.x = sS[fm][swz(fm, kk + koff)];
      a.y = sS[fm][swz(fm, kk + koff + 1)];
      const float* Vr = Vb + (size_t)(kk + koff) * D + n0 + fm;
      b.x = Vr[0];
      b.y = Vr[D];
      acc = __builtin_amdgcn_wmma_f32_16x16x4_f32(false, a, false, b,
                                                  (short)0, acc, false, false);
    }
    #pragma unroll
    for (int r = 0; r < 8; ++r) {
      const int m = half * 8 + r;
      out_result[((size_t)bh * T + q0 + m) * D + n0 + fm] = acc[r];
    }
  }
}

extern "C" void kernel_launch(void* const* d_in, const int* in_sizes, int n_in,
                              void* d_out, int out_size, void* d_ws, size_t ws_size,
                              hipStream_t stream) {
  (void)in_sizes; (void)n_in; (void)out_size; (void)d_ws; (void)ws_size;
  const float*   key   = (const float*)d_in[0];
  const float*   value = (const float*)d_in[1];
  const float*   query = (const float*)d_in[2];
  const uint8_t* mask  = (const uint8_t*)d_in[3];
  const float*   qmask = (const float*)d_in[4];
  const float*   gauss = (const float*)d_in[5];

  float* out_result = (float*)d_out;                         // [BH,T,D]
  float* out_attn   = (float*)d_out + (size_t)BH * T * D;    // [BH,T,T]

  dim3 grid(BH * (T / QT));  // 4096 blocks
  mha_fwd_kernel<<<grid, NTHREADS, 0, stream>>>(key, value, query, mask,
                                                qmask, gauss, out_result, out_attn);
}